// LayerNormLSTM_22041772163701
// MI455X (gfx1250) — compile-verified
//
#include <hip/hip_runtime.h>
#include <math.h>

// ---------------------------------------------------------------------------
// LayerNorm-LSTM (bidirectional, 2 layers) for MI455X / gfx1250.
// B=64, S=48, E=H=512, 4H=2048. fp32 math via V_WMMA_F32_16X16X4_F32.
//
// Design (reasoned from MI455X specs, compile-only):
//  * All weights + precomputed input gates (~85MB) are L2-resident (192MB);
//    HBM traffic ~100MB => ~4us @23.3TB/s. The scan is latency bound, so
//    minimize per-step sync: each WG owns a 16-row batch stripe end-to-end
//    (WMMA GEMM -> LN -> gates -> c/h update), no cross-WG dependencies.
//  * 320KB LDS/WGP lets a WG hold a full 16x2048 fp32 gate stripe (+A stripe)
//    so the gate-dimension LayerNorm is done entirely in LDS.
//  * fp32 WMMA keeps exact reference precision (LSTM recurrence compounds
//    rounding error; no need to downconvert given everything is cache-resident).
// ---------------------------------------------------------------------------

#define B_   64
#define S_   48
#define E_   512
#define H_   512
#define G4_  2048
#define EPS_ 1e-5f
#define GP_  (G4_ + 8)   // padded LDS gate-row stride (floats) to avoid 16-way bank conflicts
#define APAD_ 8          // padded LDS A-row extra floats

typedef float v2f __attribute__((ext_vector_type(2)));
typedef float v8f __attribute__((ext_vector_type(8)));

__device__ __forceinline__ float waveRedSum(float v) {
#pragma unroll
  for (int off = 16; off > 0; off >>= 1) v += __shfl_xor(v, off, 32);
  return v;
}

// Compute ldsG[16][G4_] = ldsA[16][K] * Wd[G4_][K]^T  (C[m,n] = sum_k A[m,k]*W[n,k])
// 8 waves; each wave owns 16 n-tiles of 16x16, K-loop in steps of 4 via
// V_WMMA_F32_16X16X4_F32.
//   A 16x4 f32 layout: lane L -> row (L&15), K = 2*(L>>4) + {0,1}
//   B  4x16 f32 layout: lane L -> col (L&15), K = 2*(L>>4) + {0,1}
//   C/D 16x16 f32: VGPR v -> row v + 8*(L>>4), col (L&15)
__device__ __forceinline__ void gemm_tile_rows16(
    const float* __restrict__ ldsA, int K,
    const float* __restrict__ Wd,   // [G4_][K], row-major
    float* __restrict__ ldsG)
{
  const int tid  = threadIdx.x;
  const int wave = tid >> 5;
  const int lane = tid & 31;
  const int nl   = lane & 15;
  const int kb   = (lane >> 4) << 1;        // 0 or 2
  const float* arow = ldsA + nl * (K + APAD_) + kb;
  const int rbase = (lane >> 4) << 3;       // 0 or 8

#pragma unroll 1
  for (int it = 0; it < 16; ++it) {
    const int n0 = wave * 256 + it * 16;
    const float* wrow = Wd + (size_t)(n0 + nl) * K + kb;
    v8f acc = {0.f, 0.f, 0.f, 0.f, 0.f, 0.f, 0.f, 0.f};
#pragma unroll 4
    for (int k0 = 0; k0 < K; k0 += 4) {
      v2f a = *(const v2f*)(arow + k0);
      v2f b = *(const v2f*)(wrow + k0);
      acc = __builtin_amdgcn_wmma_f32_16x16x4_f32(
          /*neg_a=*/false, a, /*neg_b=*/false, b,
          /*c_mod=*/(short)0, acc, /*reuse_a=*/false, /*reuse_b=*/false);
    }
#pragma unroll
    for (int v = 0; v < 8; ++v)
      ldsG[(rbase + v) * GP_ + n0 + nl] = acc[v];
  }
}

// OUT[dir][r][0:G4] = LN( A[r,:] @ W[dir]^T , lng[dir], lnb[dir] )
// Amode 0: A is [M][K] row-major.  Amode 1: A is x[B][S][E], r = s*B + b.
__global__ void gemm_ln_kernel(
    const float* __restrict__ A, int Amode, int K,
    const float* __restrict__ W,    // [2][G4_][K]
    const float* __restrict__ lng,  // [2][G4_]
    const float* __restrict__ lnb,  // [2][G4_]
    float* __restrict__ out,        // [2][M][G4_]
    int M)
{
  extern __shared__ float smem[];
  float* ldsA = smem;
  float* ldsG = smem + 16 * (K + APAD_);
  const int tid    = threadIdx.x;
  const int stripe = blockIdx.x;
  const int dir    = blockIdx.y;
  const int r0     = stripe * 16;

  // Stage 16 A rows into LDS (coalesced float4 per 16-thread group)
  {
    const int row  = tid >> 4;
    const int tsub = tid & 15;
    const int r    = r0 + row;
    const float* grow;
    if (Amode == 1) {
      const int s = r >> 6;   // r / B_
      const int b = r & 63;   // r % B_
      grow = A + ((size_t)b * S_ + s) * E_;
    } else {
      grow = A + (size_t)r * K;
    }
    float* lrow = ldsA + row * (K + APAD_);
    const int nv = K >> 2;
    for (int idx = tsub; idx < nv; idx += 16)
      ((float4*)lrow)[idx] = ((const float4*)grow)[idx];
  }
  __syncthreads();

  gemm_tile_rows16(ldsA, K, W + (size_t)dir * G4_ * K, ldsG);
  __syncthreads();

  const int wave = tid >> 5, lane = tid & 31;
  const float* g  = lng + dir * G4_;
  const float* bt = lnb + dir * G4_;
#pragma unroll 1
  for (int rr = 0; rr < 2; ++rr) {
    const int row = wave * 2 + rr;
    const float* grow = ldsG + row * GP_;
    float sum = 0.f, sq = 0.f;
#pragma unroll 4
    for (int j = 0; j < 64; ++j) {
      float v = grow[lane + 32 * j];
      sum += v; sq += v * v;
    }
    sum = waveRedSum(sum); sq = waveRedSum(sq);
    float mu = sum * (1.f / G4_);
    float rs = rsqrtf(sq * (1.f / G4_) - mu * mu + EPS_);
    float* orow = out + ((size_t)dir * M + (r0 + row)) * G4_;
#pragma unroll 4
    for (int j = 0; j < 64; ++j) {
      int col = lane + 32 * j;
      float v = grow[col];
      orow[col] = (v - mu) * rs * g[col] + bt[col];
    }
  }
}

// One LSTM time-step for both directions; each WG owns a 16-row batch stripe.
// gih row address = gih + dir*strideDir + t*strideT + b*G4_ (strideT=0 => layer1).
__global__ void lstm_step_kernel(
    const float* __restrict__ gih, long strideDir, long strideT,
    const float* __restrict__ Whh,   // [2][G4_][H_]
    const float* __restrict__ lnhg, const float* __restrict__ lnhb, // [2][G4_]
    const float* __restrict__ lnog, const float* __restrict__ lnob, // [2][H_]
    float* __restrict__ hbuf, float* __restrict__ cbuf,             // [2][B_][H_]
    int s)
{
  extern __shared__ float smem[];
  float* ldsA = smem;
  float* ldsG = smem + 16 * (H_ + APAD_);
  const int tid    = threadIdx.x;
  const int stripe = blockIdx.x;   // 0..3
  const int dir    = blockIdx.y;   // 0..1
  const int b0     = stripe * 16;
  const int t      = (dir == 0) ? s : (S_ - 1 - s);

  // Stage 16 h rows (this WG's own rows; updated in place at the end)
  {
    const int row  = tid >> 4;
    const int tsub = tid & 15;
    const float* grow = hbuf + ((size_t)dir * B_ + b0 + row) * H_;
    float* lrow = ldsA + row * (H_ + APAD_);
    for (int idx = tsub; idx < (H_ >> 2); idx += 16)
      ((float4*)lrow)[idx] = ((const float4*)grow)[idx];
  }
  __syncthreads();

  gemm_tile_rows16(ldsA, H_, Whh + (size_t)dir * G4_ * H_, ldsG);
  __syncthreads();

  const int wave = tid >> 5, lane = tid & 31;
  const float* hg = lnhg + dir * G4_;
  const float* hb = lnhb + dir * G4_;
  const float* og = lnog + dir * H_;
  const float* ob = lnob + dir * H_;

#pragma unroll 1
  for (int rr = 0; rr < 2; ++rr) {
    const int row = wave * 2 + rr;
    const int b   = b0 + row;
    float* grow = ldsG + row * GP_;
    const float* girow =
        gih + (size_t)dir * strideDir + (size_t)t * strideT + (size_t)b * G4_;

    // LN over the 2048 recurrent pre-activations, add input gates, activate
    float sum = 0.f, sq = 0.f;
#pragma unroll 4
    for (int j = 0; j < 64; ++j) {
      float v = grow[lane + 32 * j];
      sum += v; sq += v * v;
    }
    sum = waveRedSum(sum); sq = waveRedSum(sq);
    float mu = sum * (1.f / G4_);
    float rs = rsqrtf(sq * (1.f / G4_) - mu * mu + EPS_);
#pragma unroll 1
    for (int j = 0; j < 64; ++j) {
      int col = lane + 32 * j;
      float v = (grow[col] - mu) * rs * hg[col] + hb[col] + girow[col];
      float act = (col < 3 * H_) ? (1.f / (1.f + __expf(-v))) : tanhf(v);
      grow[col] = act;
    }

    // c = f*c + i*g; h = o * tanh(LN(c))
    float* crow = cbuf + ((size_t)dir * B_ + b) * H_;
    float* hrow = hbuf + ((size_t)dir * B_ + b) * H_;
    float cs = 0.f, cq = 0.f;
    float cn[16], ov[16];
#pragma unroll
    for (int j = 0; j < 16; ++j) {
      int col = lane + 32 * j;
      float i_ = grow[col];
      float f_ = grow[H_ + col];
      float o_ = grow[2 * H_ + col];
      float g_ = grow[3 * H_ + col];
      float c2 = f_ * crow[col] + i_ * g_;
      crow[col] = c2;
      cn[j] = c2; ov[j] = o_;
      cs += c2; cq += c2 * c2;
    }
    cs = waveRedSum(cs); cq = waveRedSum(cq);
    float mu2 = cs * (1.f / H_);
    float rs2 = rsqrtf(cq * (1.f / H_) - mu2 * mu2 + EPS_);
#pragma unroll
    for (int j = 0; j < 16; ++j) {
      int col = lane + 32 * j;
      hrow[col] = ov[j] * tanhf((cn[j] - mu2) * rs2 * og[col] + ob[col]);
    }
  }
}

__global__ void zero_kernel(float* __restrict__ p, int n) {
  int i = blockIdx.x * blockDim.x + threadIdx.x;
  if (i < n) p[i] = 0.f;
}

// dst[b][dir*H + col] = src[(dir*B + b)*H + col]   (used for xc and final output)
__global__ void interleave_kernel(const float* __restrict__ src, float* __restrict__ dst) {
  int i = blockIdx.x * blockDim.x + threadIdx.x;   // over 2*B_*H_
  if (i < 2 * B_ * H_) {
    int col = i & (H_ - 1);
    int b   = (i >> 9) & 63;
    int dir = i >> 15;
    dst[((size_t)b * 2 + dir) * H_ + col] = src[i];
  }
}

extern "C" void kernel_launch(void* const* d_in, const int* in_sizes, int n_in,
                              void* d_out, int out_size, void* d_ws, size_t ws_size,
                              hipStream_t stream) {
  const float* x        = (const float*)d_in[0];
  // d_in[1] = text_length (unused by the reference forward)
  const float* w_ih0    = (const float*)d_in[2];
  const float* w_hh0    = (const float*)d_in[3];
  const float* ln_ih0_g = (const float*)d_in[4];
  const float* ln_ih0_b = (const float*)d_in[5];
  const float* ln_hh0_g = (const float*)d_in[6];
  const float* ln_hh0_b = (const float*)d_in[7];
  const float* ln_ho0_g = (const float*)d_in[8];
  const float* ln_ho0_b = (const float*)d_in[9];
  const float* w_ih1    = (const float*)d_in[10];
  const float* w_hh1    = (const float*)d_in[11];
  const float* ln_ih1_g = (const float*)d_in[12];
  const float* ln_ih1_b = (const float*)d_in[13];
  const float* ln_hh1_g = (const float*)d_in[14];
  const float* ln_hh1_b = (const float*)d_in[15];
  const float* ln_ho1_g = (const float*)d_in[16];
  const float* ln_ho1_b = (const float*)d_in[17];
  float* out = (float*)d_out;

  // Workspace layout (floats): ~52.5 MB total
  float* ws   = (float*)d_ws;
  float* gih0 = ws;                                  // [2][S][B][G4]
  float* g1   = gih0 + (size_t)2 * S_ * B_ * G4_;    // [2][B][G4]
  float* h0   = g1 + (size_t)2 * B_ * G4_;           // [2][B][H]
  float* c0   = h0 + (size_t)2 * B_ * H_;
  float* h1   = c0 + (size_t)2 * B_ * H_;
  float* c1   = h1 + (size_t)2 * B_ * H_;
  float* xc   = c1 + (size_t)2 * B_ * H_;            // [B][2H]

  // Zero all four state buffers (contiguous: h0,c0,h1,c1)
  {
    int nz = 4 * 2 * B_ * H_;
    zero_kernel<<<(nz + 255) / 256, 256, 0, stream>>>(h0, nz);
  }

  // Layer 0 input gates: LN(x @ W_ih0^T) for both directions
  {
    size_t sh = (size_t)(16 * (E_ + APAD_) + 16 * GP_) * sizeof(float);
    gemm_ln_kernel<<<dim3(S_ * B_ / 16, 2), 256, sh, stream>>>(
        x, /*Amode=*/1, E_, w_ih0, ln_ih0_g, ln_ih0_b, gih0, S_ * B_);
  }

  // Layer 0 scan (fwd reads t=s, bwd reads t=S-1-s inside the kernel)
  size_t shstep = (size_t)(16 * (H_ + APAD_) + 16 * GP_) * sizeof(float);
  for (int s = 0; s < S_; ++s) {
    lstm_step_kernel<<<dim3(4, 2), 256, shstep, stream>>>(
        gih0, (long)S_ * B_ * G4_, (long)B_ * G4_,
        w_hh0, ln_hh0_g, ln_hh0_b, ln_ho0_g, ln_ho0_b, h0, c0, s);
  }

  // xc = concat(h_fwd, h_bwd) along feature dim
  interleave_kernel<<<(2 * B_ * H_ + 255) / 256, 256, 0, stream>>>(h0, xc);

  // Layer 1 input gates: LN(xc @ W_ih1^T), constant over time (aliasing bug)
  {
    size_t sh = (size_t)(16 * (2 * H_ + APAD_) + 16 * GP_) * sizeof(float);
    gemm_ln_kernel<<<dim3(B_ / 16, 2), 256, sh, stream>>>(
        xc, /*Amode=*/0, 2 * H_, w_ih1, ln_ih1_g, ln_ih1_b, g1, B_);
  }

  // Layer 1 scan with constant input gates (strideT = 0)
  for (int s = 0; s < S_; ++s) {
    lstm_step_kernel<<<dim3(4, 2), 256, shstep, stream>>>(
        g1, (long)B_ * G4_, 0L,
        w_hh1, ln_hh1_g, ln_hh1_b, ln_ho1_g, ln_ho1_b, h1, c1, s);
  }

  // out[b] = concat(hf1[b], hb1[b])
  interleave_kernel<<<(2 * B_ * H_ + 255) / 256, 256, 0, stream>>>(h1, out);
}